// Adapt_CSA_57028575756395
// MI455X (gfx1250) — compile-verified
//
#include <hip/hip_runtime.h>
#include <stdint.h>

// Shapes fixed by the reference: x (32, 256, 64, 64) fp32.
#define B_   32
#define C_   256
#define H_   64
#define W_   64
#define HW_  4096           // H*W
#define CHW_ (C_ * HW_)     // 1048576

typedef float fvec4 __attribute__((ext_vector_type(4)));

__device__ __forceinline__ float sigmoidf(float v) {
    return 1.0f / (1.0f + __expf(-v));
}

// ---------------------------------------------------------------------------
// Kernel 1: global average pool over H*W for each (b,c).
// One 256-thread block per (b,c); each thread sums 4 float4s (16 floats).
// ---------------------------------------------------------------------------
__global__ void k_pool(const float* __restrict__ x, float* __restrict__ pool) {
    const int bc  = blockIdx.x;     // 0 .. B_*C_-1
    const int tid = threadIdx.x;    // 0 .. 255
    const fvec4* xp = (const fvec4*)(x + (size_t)bc * HW_);
    float s = 0.0f;
#pragma unroll
    for (int i = 0; i < 4; ++i) {
        fvec4 v = xp[tid + i * 256];
        s += v.x + v.y + v.z + v.w;
    }
    __shared__ float red[256];
    red[tid] = s;
    __syncthreads();
    for (int off = 128; off > 0; off >>= 1) {
        if (tid < off) red[tid] += red[tid + off];
        __syncthreads();
    }
    if (tid == 0) pool[bc] = red[0] * (1.0f / (float)HW_);
}

// ---------------------------------------------------------------------------
// Kernel 2: 1D cross-correlation (XLA conv) over the channel dim, SAME pad.
// y[b,c] = sum_{k=0..4} w_c[k] * pool[b, c+k-2]   (zero padded)
// ---------------------------------------------------------------------------
__global__ void k_conv1d(const float* __restrict__ pool,
                         const float* __restrict__ wc,
                         float* __restrict__ y) {
    const int idx = blockIdx.x * 256 + threadIdx.x;  // 0 .. B_*C_-1
    const int b = idx >> 8;
    const int c = idx & (C_ - 1);
    const float* pb = pool + b * C_;
    float acc = 0.0f;
#pragma unroll
    for (int k = 0; k < 5; ++k) {
        const int cc = c + k - 2;
        if (cc >= 0 && cc < C_) acc += wc[k] * pb[cc];
    }
    y[idx] = acc;
}

// ---------------------------------------------------------------------------
// Kernel 3: per-pixel channel partial sum & max of sigmoid(x * y).
// Grid: 32 batches x 4 pixel-tiles (1024 px) x 2 channel halves = 256 blocks.
// Each lane owns a private 16-byte LDS slot per buffer and streams its 4
// pixels through a 4-deep global_load_async_to_lds_b128 pipeline:
//   - steady state keeps 4 loads in flight per wave (s_wait_asynccnt 3)
//   - s_wait_dscnt 0 before each issue guarantees the buffer being
//     overwritten has been fully read back (no block barrier needed since
//     lanes only read the LDS bytes their own async op wrote).
// ---------------------------------------------------------------------------
__global__ void k_stats(const float* __restrict__ x,
                        const float* __restrict__ y,
                        float* __restrict__ sump,
                        float* __restrict__ maxp) {
    const int tid  = threadIdx.x;            // 0..255
    const int b    = blockIdx.x >> 3;
    const int t    = (blockIdx.x >> 1) & 3;  // pixel tile of 1024 pixels
    const int half = blockIdx.x & 1;         // channel half
    const int c0   = half * 128;

    __shared__ float ybuf[128];
    __shared__ float xbuf[4][1024];          // 16 KB, 4-deep pipeline

    if (tid < 128) ybuf[tid] = y[b * C_ + c0 + tid];
    __syncthreads();

    const int p = t * 1024 + tid * 4;        // this thread's 4 pixels
    const unsigned long long g0 = (unsigned long long)(uintptr_t)
        (x + (size_t)b * CHW_ + (size_t)c0 * HW_ + p);
    const unsigned lds_base = (unsigned)(uintptr_t)(&xbuf[0][0]) + tid * 16;

    // prime the pipeline: channels c0+0..2 -> buffers 0..2
#pragma unroll
    for (int i = 0; i < 3; ++i) {
        const unsigned l = lds_base + ((unsigned)i << 12);
        const unsigned long long g = g0 + (unsigned long long)i * (HW_ * 4);
        asm volatile("global_load_async_to_lds_b128 %0, %1, off"
                     :: "v"(l), "v"(g) : "memory");
    }

    fvec4 sum = {0.0f, 0.0f, 0.0f, 0.0f};
    fvec4 mx  = {-3.402823466e38f, -3.402823466e38f,
                 -3.402823466e38f, -3.402823466e38f};

    auto consume = [&](int i) {
        const fvec4 v  = *(const fvec4*)(&xbuf[i & 3][tid * 4]);
        const float yc = ybuf[i];
        fvec4 s;
        s.x = sigmoidf(v.x * yc);
        s.y = sigmoidf(v.y * yc);
        s.z = sigmoidf(v.z * yc);
        s.w = sigmoidf(v.w * yc);
        sum += s;
        mx.x = fmaxf(mx.x, s.x);
        mx.y = fmaxf(mx.y, s.y);
        mx.z = fmaxf(mx.z, s.z);
        mx.w = fmaxf(mx.w, s.w);
    };

    for (int i = 0; i < 125; ++i) {
        // ensure the buffer we are about to overwrite has been read back
        asm volatile("s_wait_dscnt 0x0" ::: "memory");
        const unsigned l = lds_base + ((unsigned)((i + 3) & 3) << 12);
        const unsigned long long g =
            g0 + (unsigned long long)(i + 3) * (HW_ * 4);
        asm volatile("global_load_async_to_lds_b128 %0, %1, off"
                     :: "v"(l), "v"(g) : "memory");
        // 4 in flight; loads complete in order -> <=3 means channel i landed
        asm volatile("s_wait_asynccnt 0x3" ::: "memory");
        consume(i);
    }
    asm volatile("s_wait_asynccnt 0x2" ::: "memory"); consume(125);
    asm volatile("s_wait_asynccnt 0x1" ::: "memory"); consume(126);
    asm volatile("s_wait_asynccnt 0x0" ::: "memory"); consume(127);

    const size_t o = (size_t)half * (B_ * HW_) + b * HW_ + p;
    *(fvec4*)(sump + o) = sum;
    *(fvec4*)(maxp + o) = mx;
}

// ---------------------------------------------------------------------------
// Kernel 3b: combine the two channel-half partials into mean & max.
// ---------------------------------------------------------------------------
__global__ void k_combine(const float* __restrict__ sump,
                          const float* __restrict__ maxp,
                          float* __restrict__ avg,
                          float* __restrict__ mxo) {
    const int i = blockIdx.x * 256 + threadIdx.x;   // fvec4 index < 32768
    const fvec4 s0 = ((const fvec4*)sump)[i];
    const fvec4 s1 = ((const fvec4*)sump)[i + (B_ * HW_ / 4)];
    const fvec4 m0 = ((const fvec4*)maxp)[i];
    const fvec4 m1 = ((const fvec4*)maxp)[i + (B_ * HW_ / 4)];
    const fvec4 a = (s0 + s1) * (1.0f / (float)C_);
    fvec4 m;
    m.x = fmaxf(m0.x, m1.x);
    m.y = fmaxf(m0.y, m1.y);
    m.z = fmaxf(m0.z, m1.z);
    m.w = fmaxf(m0.w, m1.w);
    ((fvec4*)avg)[i] = a;
    ((fvec4*)mxo)[i] = m;
}

// ---------------------------------------------------------------------------
// Kernel 4: 5x5 SAME cross-correlation over 2 channels [avg, max] + sigmoid.
// One thread per output pixel (B*H*W = 131072).
// w_s layout (1,2,5,5): channel 0 taps = w[0..24], channel 1 taps = w[25..49].
// ---------------------------------------------------------------------------
__global__ void k_conv2d(const float* __restrict__ avg,
                         const float* __restrict__ mxo,
                         const float* __restrict__ ws5,
                         float* __restrict__ sa) {
    __shared__ float w[50];
    const int tid = threadIdx.x;
    if (tid < 50) w[tid] = ws5[tid];
    __syncthreads();

    const int idx = blockIdx.x * 256 + tid;   // 0 .. B_*HW_-1
    const int b = idx >> 12;
    const int p = idx & (HW_ - 1);
    const int h = p >> 6;
    const int wc = p & 63;
    const float* a = avg + b * HW_;
    const float* m = mxo + b * HW_;

    float acc = 0.0f;
#pragma unroll
    for (int kh = 0; kh < 5; ++kh) {
        const int hh = h + kh - 2;
        if (hh < 0 || hh >= H_) continue;
#pragma unroll
        for (int kw = 0; kw < 5; ++kw) {
            const int ww = wc + kw - 2;
            if (ww < 0 || ww >= W_) continue;
            const int q = hh * W_ + ww;
            acc += w[kh * 5 + kw] * a[q] + w[25 + kh * 5 + kw] * m[q];
        }
    }
    sa[idx] = sigmoidf(acc);
}

// ---------------------------------------------------------------------------
// Kernel 5: out = sigmoid(x*y) * sa + x, fully vectorized (float4).
// Output is never re-read -> non-temporal stores keep x resident in the
// 192 MB L2 across passes (x itself is only 128 MB).
// ---------------------------------------------------------------------------
__global__ void k_fuse(const float* __restrict__ x,
                       const float* __restrict__ y,
                       const float* __restrict__ sa,
                       float* __restrict__ out) {
    const fvec4* x4 = (const fvec4*)x;
    const fvec4* s4 = (const fvec4*)sa;
    fvec4* o4 = (fvec4*)out;
#pragma unroll
    for (int k = 0; k < 4; ++k) {
        const int idx = blockIdx.x * 1024 + k * 256 + threadIdx.x; // float4 id
        const int bc = idx >> 10;           // 1024 float4 per (b,c) plane
        const int b  = bc >> 8;
        const int p4 = idx & 1023;
        const fvec4 xv = x4[idx];
        const float yv = y[bc];
        const fvec4 sv = s4[(b << 10) + p4];
        const fvec4 tv = xv * yv;
        fvec4 sg;
        sg.x = sigmoidf(tv.x);
        sg.y = sigmoidf(tv.y);
        sg.z = sigmoidf(tv.z);
        sg.w = sigmoidf(tv.w);
        const fvec4 r = sg * sv + xv;
        __builtin_nontemporal_store(r, &o4[idx]);
    }
}

// ---------------------------------------------------------------------------
extern "C" void kernel_launch(void* const* d_in, const int* in_sizes, int n_in,
                              void* d_out, int out_size, void* d_ws, size_t ws_size,
                              hipStream_t stream) {
    (void)in_sizes; (void)n_in; (void)out_size; (void)ws_size;

    const float* x   = (const float*)d_in[0];   // (32,256,64,64)
    const float* wc  = (const float*)d_in[1];   // (1,1,5)
    const float* ws5 = (const float*)d_in[2];   // (1,2,5,5)
    float* out = (float*)d_out;

    // Workspace layout (floats):
    //   pool 8192 | y 8192 | sump 262144 | maxp 262144 |
    //   avg 131072 | mxo 131072 | sa 131072          (~3.7 MB total)
    float* pool = (float*)d_ws;
    float* y    = pool + B_ * C_;
    float* sump = y    + B_ * C_;
    float* maxp = sump + 2 * B_ * HW_;
    float* avg  = maxp + 2 * B_ * HW_;
    float* mxo  = avg  + B_ * HW_;
    float* sa   = mxo  + B_ * HW_;

    k_pool   <<<B_ * C_,            256, 0, stream>>>(x, pool);
    k_conv1d <<<(B_ * C_) / 256,    256, 0, stream>>>(pool, wc, y);
    k_stats  <<<B_ * 8,             256, 0, stream>>>(x, y, sump, maxp);
    k_combine<<<(B_ * HW_ / 4) / 256, 256, 0, stream>>>(sump, maxp, avg, mxo);
    k_conv2d <<<(B_ * HW_) / 256,   256, 0, stream>>>(avg, mxo, ws5, sa);
    k_fuse   <<<(B_ * CHW_) / 4096, 256, 0, stream>>>(x, y, sa, out);
}